// Model_7481833030317
// MI455X (gfx1250) — compile-verified
//
#include <hip/hip_runtime.h>
#include <stdint.h>

// ---------------------------------------------------------------------------
// MI455X / gfx1250, wave32. fp32 network -> V_WMMA_F32_16X16X4_F32 everywhere.
// All conv/GEMM inner loops: 1x global_load_b64 (A) + 1x ds_load_b64 (B) +
// 1x v_wmma per K-step of 4; weights repacked once per call into (co,tap,ci)
// order so A pairs are contiguous; LDS B tiles are channel-innermost with
// +4 channel padding (bank spread + 8B alignment).
// ---------------------------------------------------------------------------

typedef __attribute__((ext_vector_type(2))) float v2f;
typedef __attribute__((ext_vector_type(8))) float v8f;

__device__ __forceinline__ v8f wmma_f32(v2f a, v2f b, v8f c) {
  return __builtin_amdgcn_wmma_f32_16x16x4_f32(false, a, false, b, (short)0, c,
                                               false, false);
}

// ---------------------------------------------------------------------------
// Weight repack: src OIHW-style [co][ci][tap] -> dst [co][tap][cip], zero pad
// channels cip in [Ci, CIP). Also used for the SCNN kernels (tap == j).
// ---------------------------------------------------------------------------
extern "C" __global__ void repack_w(const float* __restrict__ src,
                                    float* __restrict__ dst, int Co, int Ci,
                                    int CIP) {
  const int total = Co * 9 * CIP;
  for (int i = blockIdx.x * blockDim.x + threadIdx.x; i < total;
       i += gridDim.x * blockDim.x) {
    int cip = i % CIP;
    int t = i / CIP;
    int tap = t % 9;
    int co = t / 9;
    dst[i] = (cip < Ci) ? src[(co * Ci + cip) * 9 + tap] : 0.f;
  }
}

// ---------------------------------------------------------------------------
// Fused 3x3 conv (pad == DIL) + optional bias + scale/shift + ReLU.
// One wave per 16-wide Co tile, block covers all CO for one 16-pixel strip.
// Epilogue: stage 16x16 tile in LDS, drain with async LDS->global b128.
// ---------------------------------------------------------------------------
template <int CI, int CIR, int CO, int DIL>
__global__ __launch_bounds__(CO * 2) void conv3x3_t(
    const float* __restrict__ in, const float* __restrict__ WR,
    const float* __restrict__ bias, const float* __restrict__ scale,
    const float* __restrict__ shift, float* __restrict__ out, int H, int W) {
  constexpr int SW = 16 + 2 * DIL;   // staged row width
  constexpr int CP = CI + 4;         // channel pitch (bank spread, even)
  constexpr int K = 9 * CI;
  constexpr int NWAVE = CO / 16;
  __shared__ float smem[3 * SW * CP + NWAVE * 256];

  const int ow0 = blockIdx.x * 16;
  const int oh = blockIdx.y;
  const int nb = blockIdx.z;
  const int tid = threadIdx.x;
  float* sIn = smem;  // [(r*SW + col)*CP + ci]

  const float* inN = in + (size_t)nb * CIR * H * W;
  for (int idx = tid; idx < 3 * SW * CI; idx += CO * 2) {
    int c = idx % SW;
    int t = idx / SW;
    int r = t % 3;
    int ci = t / 3;
    int ih = oh + (r - 1) * DIL;
    int iw = ow0 + c - DIL;
    float v = 0.f;
    if (ci < CIR && ih >= 0 && ih < H && iw >= 0 && iw < W)
      v = inN[((size_t)ci * H + ih) * W + iw];
    sIn[(r * SW + c) * CP + ci] = v;
  }
  __syncthreads();

  const int wave = tid >> 5;
  const int lane = tid & 31;
  const int m = lane & 15;
  const int half = (lane >> 4) & 1;  // clamp range for the optimizer
  const int co0 = wave * 16;

  const float* Arow = WR + (size_t)(co0 + m) * K + 2 * half;
  __builtin_prefetch(Arow, 0, 0);  // global_prefetch_b8

  v8f acc = {0.f, 0.f, 0.f, 0.f, 0.f, 0.f, 0.f, 0.f};
#pragma unroll
  for (int tap = 0; tap < 9; ++tap) {
    constexpr int KH[9] = {0, 0, 0, 1, 1, 1, 2, 2, 2};
    constexpr int KW[9] = {0, 1, 2, 0, 1, 2, 0, 1, 2};
    const float* aB = Arow + tap * CI;
    const float* bB = &sIn[(KH[tap] * SW + m + KW[tap] * DIL) * CP + 2 * half];
#pragma unroll 4
    for (int ci = 0; ci < CI; ci += 4) {
      v2f a = *(const v2f*)(aB + ci);
      v2f b = *(const v2f*)(bB + ci);
      acc = wmma_f32(a, b, acc);
    }
  }

  // Epilogue: BN + ReLU into a per-wave LDS tile [co_local][16].
  float* sOut = smem + 3 * SW * CP + wave * 256;
#pragma unroll
  for (int r = 0; r < 8; ++r) {
    int col = r + 8 * half;
    int co = co0 + col;
    float v = acc[r];
    if (bias) v += bias[co];
    v = fmaxf(v * scale[co] + shift[co], 0.f);
    sOut[col * 16 + m] = v;
  }
  asm volatile("s_wait_dscnt 0" ::: "memory");

  // Drain tile: 16 rows x 64B, coalesced async LDS->global stores.
  float* outN = out + (size_t)nb * CO * H * W;
#pragma unroll
  for (int i = 0; i < 2; ++i) {
    int chunk = lane + 32 * i;  // 0..63
    int row = chunk >> 2;       // co_local
    int part = chunk & 3;       // 16B quarter of the row
    uint32_t lsrc = (uint32_t)(uintptr_t)(sOut + row * 16 + part * 4);
    const float* gdst =
        outN + ((size_t)(co0 + row) * H + oh) * W + ow0 + part * 4;
    asm volatile("global_store_async_from_lds_b128 %0, %1, off" ::"v"(
                     (uint64_t)(uintptr_t)gdst),
                 "v"(lsrc)
                 : "memory");
  }
  asm volatile("s_wait_asynccnt 0" ::: "memory");
}

template <int CI, int CIR, int CO, int DIL>
static void run_conv(const float* in, const float* WR, const float* b,
                     const float* s, const float* t, float* o, int H, int W,
                     hipStream_t st) {
  dim3 grid(W / 16, H, 4);
  conv3x3_t<CI, CIR, CO, DIL><<<grid, CO * 2, 0, st>>>(in, WR, b, s, t, o, H,
                                                       W);
}

// ---------------------------------------------------------------------------
// 1x1 conv (256 -> 128) + scale/shift + ReLU.  K = ci order is already
// contiguous in the raw weights, so no repack needed.
// ---------------------------------------------------------------------------
extern "C" __global__ __launch_bounds__(256) void conv1x1_bn_relu(
    const float* __restrict__ in, const float* __restrict__ Wt,
    const float* __restrict__ scale, const float* __restrict__ shift,
    float* __restrict__ out, int HW) {
  constexpr int CI = 256, CO = 128, CP = CI + 4;
  __shared__ float sIn[16 * CP];  // [pixel][ci]
  const int p0 = blockIdx.x * 16;
  const int nb = blockIdx.z;
  const int tid = threadIdx.x;
  const float* inN = in + (size_t)nb * CI * HW;
  for (int idx = tid; idx < CI * 16; idx += 256) {
    int p = idx & 15;
    int ci = idx >> 4;
    sIn[p * CP + ci] = inN[(size_t)ci * HW + p0 + p];
  }
  __syncthreads();

  const int wave = tid >> 5;
  const int lane = tid & 31;
  const int m = lane & 15;
  const int half = (lane >> 4) & 1;
  const int co0 = wave * 16;
  const float* Arow = Wt + (size_t)(co0 + m) * CI + 2 * half;
  const float* Brow = &sIn[m * CP + 2 * half];
  __builtin_prefetch(Arow, 0, 0);

  v8f acc = {0.f, 0.f, 0.f, 0.f, 0.f, 0.f, 0.f, 0.f};
#pragma unroll 8
  for (int k0 = 0; k0 < CI; k0 += 4) {
    v2f a = *(const v2f*)(Arow + k0);
    v2f b = *(const v2f*)(Brow + k0);
    acc = wmma_f32(a, b, acc);
  }

  float* outN = out + (size_t)nb * CO * HW;
#pragma unroll
  for (int r = 0; r < 8; ++r) {
    int co = co0 + r + 8 * half;
    float v = fmaxf(acc[r] * scale[co] + shift[co], 0.f);
    outN[(size_t)co * HW + p0 + m] = v;
  }
}

// ---------------------------------------------------------------------------
// 2x2 maxpool; writes the chained tensor and (optionally) a d_out feature.
// ---------------------------------------------------------------------------
extern "C" __global__ void maxpool2x2(const float* __restrict__ in,
                                      float* __restrict__ out,
                                      float* __restrict__ feat, int N, int C,
                                      int H, int W) {
  const int Ho = H >> 1, Wo = W >> 1;
  const size_t total = (size_t)N * C * Ho * Wo;
  for (size_t i = (size_t)blockIdx.x * blockDim.x + threadIdx.x; i < total;
       i += (size_t)gridDim.x * blockDim.x) {
    int w = (int)(i % Wo);
    size_t t = i / Wo;
    int h = (int)(t % Ho);
    t /= Ho;
    int c = (int)(t % C);
    int n = (int)(t / C);
    const float* p = in + (((size_t)(n * C + c) * H + 2 * h) * W + 2 * w);
    float v = fmaxf(fmaxf(p[0], p[1]), fmaxf(p[W], p[W + 1]));
    out[i] = v;
    if (feat) feat[i] = v;
  }
}

// ---------------------------------------------------------------------------
// SCNN scan pass, in place on c4 (4,128,16,32).  One workgroup (8 waves);
// previous output row cached in LDS as [batch][x+guard][ci] with 4 zero guard
// columns each side -> the k=9 window needs no bounds checks.
// out_row[s] = row[s] + relu(conv1d(out_row[s_prev]))
// ---------------------------------------------------------------------------
template <int VERT>
__global__ __launch_bounds__(256) void scnn_t(float* __restrict__ c4,
                                              const float* __restrict__ AR,
                                              int reverse) {
  constexpr int H = 16, W = 32;
  constexpr int Lscan = VERT ? H : W;
  constexpr int Lrow = VERT ? W : H;   // power of two
  constexpr int LP = Lrow + 8;
  constexpr int CP = 132;              // 128 ch + pad (bank spread, even)
  constexpr int Nt = (4 * Lrow) / 16;  // N tiles of 16 pixels
  __shared__ float sPrev[4 * LP * CP];

  const int tid = threadIdx.x;
  const int wave = tid >> 5;
  const int lane = tid & 31;
  const int m = lane & 15;
  const int half = (lane >> 4) & 1;
  const int co0 = wave * 16;

  auto elem = [](int nb, int c, int s, int x) -> size_t {
    int h = VERT ? s : x;
    int w = VERT ? x : s;
    return (((size_t)nb * 128 + c) * H + h) * W + w;
  };

  // Seed: zeros in guards/pad channels, first scan row in the interior.
  const int s0 = reverse ? (Lscan - 1) : 0;
  for (int idx = tid; idx < 4 * LP * CP; idx += 256) {
    int c = idx % CP;
    int t = idx / CP;
    int xp = t % LP;
    int nb = t / LP;
    float v = 0.f;
    if (c < 128 && xp >= 4 && xp < Lrow + 4) v = c4[elem(nb, c, s0, xp - 4)];
    sPrev[idx] = v;
  }
  __syncthreads();

  const float* Arow = AR + (size_t)(co0 + m) * 1152 + 2 * half;
  __builtin_prefetch(Arow, 0, 0);

  int pnb[Nt], px[Nt];
#pragma unroll
  for (int nt = 0; nt < Nt; ++nt) {
    int pix = nt * 16 + m;
    pnb[nt] = pix / Lrow;
    px[nt] = pix & (Lrow - 1);
  }

  for (int t = 1; t < Lscan; ++t) {
    const int s = reverse ? (Lscan - 1 - t) : t;
    v8f acc[Nt];
    const v8f vzero = {0.f, 0.f, 0.f, 0.f, 0.f, 0.f, 0.f, 0.f};
#pragma unroll
    for (int nt = 0; nt < Nt; ++nt) acc[nt] = vzero;

#pragma unroll
    for (int j = 0; j < 9; ++j) {
      const float* aB = Arow + j * 128;
#pragma unroll 4
      for (int ci = 0; ci < 128; ci += 4) {
        v2f a = *(const v2f*)(aB + ci);
#pragma unroll
        for (int nt = 0; nt < Nt; ++nt) {
          // window element x + j - 4, shifted +4 into the guard frame
          const float* bB =
              &sPrev[(pnb[nt] * LP + px[nt] + j) * CP + 2 * half];
          v2f b = *(const v2f*)(bB + ci);
          acc[nt] = wmma_f32(a, b, acc[nt]);
        }
      }
    }
    __syncthreads();  // all sPrev reads done before updates

#pragma unroll
    for (int nt = 0; nt < Nt; ++nt) {
      float vout[8];
#pragma unroll
      for (int r = 0; r < 8; ++r) {
        int co = co0 + r + 8 * half;
        size_t gi = elem(pnb[nt], co, s, px[nt]);
        float v = c4[gi] + fmaxf(acc[nt][r], 0.f);
        c4[gi] = v;
        vout[r] = v;
      }
      float* dst = &sPrev[(pnb[nt] * LP + px[nt] + 4) * CP + co0 + 8 * half];
#pragma unroll
      for (int r = 0; r < 8; r += 2) {
        v2f p2;
        p2.x = vout[r];
        p2.y = vout[r + 1];
        *(v2f*)(dst + r) = p2;  // ds_store_b64
      }
    }
    __syncthreads();  // sPrev fully updated before next step
  }
}

// ---------------------------------------------------------------------------
// Orchestration.
// ---------------------------------------------------------------------------
extern "C" void kernel_launch(void* const* d_in, const int* in_sizes, int n_in,
                              void* d_out, int out_size, void* d_ws,
                              size_t ws_size, hipStream_t stream) {
  (void)in_sizes; (void)n_in; (void)out_size; (void)ws_size;

  const float* img = (const float*)d_in[0];
  struct Blk { const float *W, *b, *s, *t; };
  Blk blk[12];
  for (int i = 0; i < 12; ++i) {
    blk[i].W = (const float*)d_in[1 + 4 * i + 0];
    blk[i].b = (const float*)d_in[1 + 4 * i + 1];
    blk[i].s = (const float*)d_in[1 + 4 * i + 2];
    blk[i].t = (const float*)d_in[1 + 4 * i + 3];
  }
  const float* l1W1 = (const float*)d_in[49];
  const float* l1s1 = (const float*)d_in[50];
  const float* l1t1 = (const float*)d_in[51];
  const float* l1W2 = (const float*)d_in[52];
  const float* l1s2 = (const float*)d_in[53];
  const float* l1t2 = (const float*)d_in[54];
  const float* ms[4] = {(const float*)d_in[55], (const float*)d_in[56],
                        (const float*)d_in[57], (const float*)d_in[58]};

  float* out = (float*)d_out;
  float* f0 = out;             // (4,16,128,256)
  float* f1 = out + 2097152;   // (4,32,64,128)
  float* f2 = out + 3145728;   // (4,64,32,64)
  float* c4 = out + 3670016;   // (4,128,16,32)

  char* ws = (char*)d_ws;
  float* bufA = (float*)ws;
  float* bufB = (float*)(ws + (size_t)(34u << 20));
  float* wreg = (float*)(ws + (size_t)(68u << 20));

  const int chans[12][2] = {{3, 16},  {16, 16},  {16, 16},  {16, 32},
                            {32, 32}, {32, 32},  {32, 64},  {64, 64},
                            {64, 64}, {64, 128}, {128, 128}, {128, 128}};

  // ---- repack all weights into (co, tap, ci) order in ws ----
  float* wr[12];
  size_t off = 0;
  for (int i = 0; i < 12; ++i) {
    int Ci = chans[i][0], Co = chans[i][1];
    int CIP = (i == 0) ? 4 : Ci;  // zero-pad layer0 channels 3 -> 4
    wr[i] = wreg + off;
    off += (size_t)Co * 9 * CIP;
    int total = Co * 9 * CIP;
    repack_w<<<(total + 255) / 256, 256, 0, stream>>>(blk[i].W, wr[i], Co, Ci,
                                                      CIP);
  }
  float* wrL1 = wreg + off;
  off += (size_t)256 * 9 * 128;
  repack_w<<<(256 * 9 * 128 + 255) / 256, 256, 0, stream>>>(l1W1, wrL1, 256,
                                                            128, 128);
  float* wrMs[4];
  for (int p = 0; p < 4; ++p) {
    wrMs[p] = wreg + off;
    off += (size_t)128 * 9 * 128;
    repack_w<<<(128 * 9 * 128 + 255) / 256, 256, 0, stream>>>(ms[p], wrMs[p],
                                                              128, 128, 128);
  }

  // ---- backbone ----
  const int Hs[4] = {256, 128, 64, 32};
  const int Wsz[4] = {512, 256, 128, 64};
  float* featDst[4] = {f0, f1, f2, nullptr};

  const float* cur = img;
  int flip = 0;
  auto nextbuf = [&]() {
    float* p = flip ? bufB : bufA;
    flip ^= 1;
    return p;
  };

  for (int st = 0; st < 4; ++st) {
    const int H = Hs[st], W = Wsz[st];
    for (int j = 0; j < 3; ++j) {
      const int bi = st * 3 + j;
      float* dst = nextbuf();
      const Blk& B = blk[bi];
      switch (bi) {
        case 0:  run_conv<4, 3, 16, 1>(cur, wr[bi], B.b, B.s, B.t, dst, H, W, stream); break;
        case 1:  run_conv<16, 16, 16, 1>(cur, wr[bi], B.b, B.s, B.t, dst, H, W, stream); break;
        case 2:  run_conv<16, 16, 16, 1>(cur, wr[bi], B.b, B.s, B.t, dst, H, W, stream); break;
        case 3:  run_conv<16, 16, 32, 1>(cur, wr[bi], B.b, B.s, B.t, dst, H, W, stream); break;
        case 4:  run_conv<32, 32, 32, 1>(cur, wr[bi], B.b, B.s, B.t, dst, H, W, stream); break;
        case 5:  run_conv<32, 32, 32, 1>(cur, wr[bi], B.b, B.s, B.t, dst, H, W, stream); break;
        case 6:  run_conv<32, 32, 64, 1>(cur, wr[bi], B.b, B.s, B.t, dst, H, W, stream); break;
        case 7:  run_conv<64, 64, 64, 1>(cur, wr[bi], B.b, B.s, B.t, dst, H, W, stream); break;
        case 8:  run_conv<64, 64, 64, 1>(cur, wr[bi], B.b, B.s, B.t, dst, H, W, stream); break;
        case 9:  run_conv<64, 64, 128, 1>(cur, wr[bi], B.b, B.s, B.t, dst, H, W, stream); break;
        case 10: run_conv<128, 128, 128, 1>(cur, wr[bi], B.b, B.s, B.t, dst, H, W, stream); break;
        default: run_conv<128, 128, 128, 1>(cur, wr[bi], B.b, B.s, B.t, dst, H, W, stream); break;
      }
      cur = dst;
    }
    const int C = chans[st * 3 + 2][1];
    float* dst = nextbuf();
    const size_t total = (size_t)4 * C * (H / 2) * (W / 2);
    maxpool2x2<<<(int)((total + 255) / 256), 256, 0, stream>>>(
        cur, dst, featDst[st], 4, C, H, W);
    cur = dst;
  }

  // Dilated 3x3 (128 -> 256, dil = 4) on (4,128,16,32)
  {
    float* dst = nextbuf();
    run_conv<128, 128, 256, 4>(cur, wrL1, nullptr, l1s1, l1t1, dst, 16, 32,
                               stream);
    cur = dst;
  }

  // 1x1 (256 -> 128) straight into the c4 output region
  {
    dim3 grid((16 * 32) / 16, 1, 4);
    conv1x1_bn_relu<<<grid, 256, 0, stream>>>(cur, l1W2, l1s2, l1t2, c4,
                                              16 * 32);
  }

  // 4 SCNN passes, in place on c4: (v,r) = (1,0),(1,1),(0,0),(0,1)
  scnn_t<1><<<1, 256, 0, stream>>>(c4, wrMs[0], 0);
  scnn_t<1><<<1, 256, 0, stream>>>(c4, wrMs[1], 1);
  scnn_t<0><<<1, 256, 0, stream>>>(c4, wrMs[2], 0);
  scnn_t<0><<<1, 256, 0, stream>>>(c4, wrMs[3], 1);
}